// CRF_67087389164150
// MI455X (gfx1250) — compile-verified
//
#include <hip/hip_runtime.h>
#include <stdint.h>

// Problem constants (from the reference): S=512, B=8192, T=7.
#define S_LEN   512
#define B_SZ    8192
#define T_TAGS  7
#define BLOCK   256
#define CHUNK   4                       // sequence steps staged per LDS buffer
#define NCHUNK  (S_LEN / CHUNK)         // 128
#define SEG_DW  (BLOCK * T_TAGS)        // 1792 dwords per step segment (7168 B)
#define CHUNK_DW (CHUNK * SEG_DW)       // 7168 dwords per chunk buffer
#define STEP_BYTES (B_SZ * T_TAGS * 4)  // 229376 B between sequence steps

typedef float v2f __attribute__((ext_vector_type(2)));

// Low 32 bits of a generic pointer to LDS == wave-relative LDS byte offset.
__device__ __forceinline__ uint32_t lds_off(const void* p) {
  return (uint32_t)(uintptr_t)p;
}

// One chunk = 4 step-segments of 7168 B. Thread group g = tid>>6 (64 threads)
// copies segment g: 448 x b128, 7 per thread. INST_OFFSET is added to both the
// LDS and the global address for async ops, so the 7 issues share registers.
__device__ __forceinline__ void issue_chunk_b128(unsigned long long gbase,
                                                 uint32_t lds0, int chunk,
                                                 int buf, int tid) {
  const int seg   = tid >> 6;           // 0..3  (uniform per wave)
  const int lane6 = tid & 63;
  const int s     = chunk * CHUNK + seg;
  const uint32_t goff = (uint32_t)s * (uint32_t)STEP_BYTES + (uint32_t)(lane6 * 16);
  const uint32_t ldst = lds0 +
      (uint32_t)((buf * CHUNK_DW + seg * SEG_DW) * 4 + lane6 * 16);
  asm volatile(
      "global_load_async_to_lds_b128 %0, %1, %2\n\t"
      "global_load_async_to_lds_b128 %0, %1, %2 offset:1024\n\t"
      "global_load_async_to_lds_b128 %0, %1, %2 offset:2048\n\t"
      "global_load_async_to_lds_b128 %0, %1, %2 offset:3072\n\t"
      "global_load_async_to_lds_b128 %0, %1, %2 offset:4096\n\t"
      "global_load_async_to_lds_b128 %0, %1, %2 offset:5120\n\t"
      "global_load_async_to_lds_b128 %0, %1, %2 offset:6144"
      :: "v"(ldst), "v"(goff), "s"(gbase) : "memory");
}

__global__ __launch_bounds__(BLOCK)
void crf_fwd_kernel(const float* __restrict__ emis,
                    const float* __restrict__ startT,
                    const float* __restrict__ endT,
                    const float* __restrict__ trans,
                    const int*   __restrict__ tags,
                    const unsigned char* __restrict__ mask,
                    float* __restrict__ partial) {
  __shared__ __align__(16) float s_emit[2][CHUNK_DW];   // 2 x 28 KB
  __shared__ float s_trans[T_TAGS * T_TAGS];
  __shared__ float s_start[T_TAGS];
  __shared__ float s_end[T_TAGS];
  __shared__ float s_red[BLOCK / 32];

  const int tid = threadIdx.x;
  const int b0  = blockIdx.x * BLOCK;
  const int b   = b0 + tid;

  if (tid < T_TAGS * T_TAGS) s_trans[tid] = trans[tid];
  if (tid < T_TAGS) { s_start[tid] = startT[tid]; s_end[tid] = endT[tid]; }

  const unsigned long long gbase =
      (unsigned long long)(uintptr_t)(emis + (size_t)b0 * T_TAGS);
  const uint32_t lds0 = lds_off(&s_emit[0][0]);

  // prime chunk 0 (7 async b128 per thread)
  issue_chunk_b128(gbase, lds0, 0, 0, tid);

  // Constant exp(transitions) table in registers, j-dimension packed in pairs
  // for V_PK_FMA_F32: er2[k][p] = {er[k][2p], er[k][2p+1]}, er6[k] = er[k][6].
  v2f   er2[T_TAGS][3];
  float er6[T_TAGS];
#pragma unroll
  for (int k = 0; k < T_TAGS; ++k) {
#pragma unroll
    for (int p = 0; p < 3; ++p) {
      er2[k][p].x = __expf(trans[k * T_TAGS + 2 * p]);
      er2[k][p].y = __expf(trans[k * T_TAGS + 2 * p + 1]);
    }
    er6[k] = __expf(trans[k * T_TAGS + 6]);
  }

  // Forward state: score[j] = Lacc + log(w[j]); renormalized once per chunk.
  v2f   wv[3];
  float w6 = 0.f;
#pragma unroll
  for (int p = 0; p < 3; ++p) wv[p] = (v2f){0.f, 0.f};
  float Lacc = 0.f;
  float num  = 0.f;
  int   prev = 0;

  for (int c = 0; c < NCHUNK; ++c) {
    const int cb = c & 1;
    if (c + 1 < NCHUNK) {
      issue_chunk_b128(gbase, lds0, c + 1, (c + 1) & 1, tid);
      asm volatile("s_wait_asynccnt 0x7" ::: "memory");  // chunk c retired
    } else {
      asm volatile("s_wait_asynccnt 0x0" ::: "memory");
    }
    __syncthreads();   // buffer cb fully populated for all waves

    // hoisted per-chunk tag/mask loads (two clauses of 4 coalesced loads)
    int      tg[CHUNK];
    unsigned mk[CHUNK];
#pragma unroll
    for (int si = 0; si < CHUNK; ++si)
      tg[si] = tags[(size_t)(c * CHUNK + si) * B_SZ + b];
#pragma unroll
    for (int si = 0; si < CHUNK; ++si)
      mk[si] = mask[(size_t)(c * CHUNK + si) * B_SZ + b];

    const float* eb = &s_emit[cb][0];
#pragma unroll
    for (int si = 0; si < CHUNK; ++si) {
      const float* ep = eb + si * SEG_DW + tid * T_TAGS;  // stride-7: no conflicts
      float e[T_TAGS];
#pragma unroll
      for (int j = 0; j < T_TAGS; ++j) e[j] = ep[j];

      const int tag = tg[si];

      if (c == 0 && si == 0) {
        // step 0: score0 = start + e0 (normalized linear form)
        float sc[T_TAGS];
        float mx = s_start[0] + e[0];
        sc[0] = mx;
#pragma unroll
        for (int j = 1; j < T_TAGS; ++j) {
          sc[j] = s_start[j] + e[j];
          mx = fmaxf(mx, sc[j]);
        }
#pragma unroll
        for (int p = 0; p < 3; ++p) {
          wv[p].x = __expf(sc[2 * p] - mx);
          wv[p].y = __expf(sc[2 * p + 1] - mx);
        }
        w6   = __expf(sc[6] - mx);
        Lacc = mx;
        num  = s_start[tag] + ep[tag];
        prev = tag;
      } else {
        const unsigned m = mk[si];
        // numerator (mask-predicated)
        const float tsc  = s_trans[prev * T_TAGS + tag];
        const float etag = ep[tag];
        num  = m ? (num + tsc + etag) : num;
        prev = m ? (int)tag : prev;

        // denominator step: d = (w . ER) * exp(e), packed over j pairs
        const float wk[T_TAGS] = {wv[0].x, wv[0].y, wv[1].x, wv[1].y,
                                  wv[2].x, wv[2].y, w6};
        v2f   dv[3];
        float d6;
        {
          const v2f s0 = {wk[0], wk[0]};
#pragma unroll
          for (int p = 0; p < 3; ++p) dv[p] = s0 * er2[0][p];
          d6 = wk[0] * er6[0];
        }
#pragma unroll
        for (int k = 1; k < T_TAGS; ++k) {
          const v2f sk = {wk[k], wk[k]};   // -> VOP3P OPSEL (lo,lo) source
#pragma unroll
          for (int p = 0; p < 3; ++p)
            dv[p] = __builtin_elementwise_fma(sk, er2[k][p], dv[p]);
          d6 = __builtin_fmaf(wk[k], er6[k], d6);
        }
        const v2f ee0 = {__expf(e[0]), __expf(e[1])};
        const v2f ee1 = {__expf(e[2]), __expf(e[3])};
        const v2f ee2 = {__expf(e[4]), __expf(e[5])};
        dv[0] = dv[0] * ee0;               // -> v_pk_mul_f32
        dv[1] = dv[1] * ee1;
        dv[2] = dv[2] * ee2;
        d6    = d6 * __expf(e[6]);
        // mask-predicated commit (no per-step renorm)
#pragma unroll
        for (int p = 0; p < 3; ++p) wv[p] = m ? dv[p] : wv[p];
        w6 = m ? d6 : w6;
      }
    }

    // once-per-chunk renormalization (identity transform on represented score;
    // growth over 4 steps <= ~2e13 from max 1.0 -- safely inside f32 range)
    {
      float mx = fmaxf(fmaxf(fmaxf(wv[0].x, wv[0].y), fmaxf(wv[1].x, wv[1].y)),
                       fmaxf(fmaxf(wv[2].x, wv[2].y), w6));
      const float inv = __builtin_amdgcn_rcpf(mx);
#pragma unroll
      for (int p = 0; p < 3; ++p) wv[p] = wv[p] * inv;
      w6 *= inv;
      Lacc += __logf(mx);
    }
    __syncthreads();   // all reads of buffer cb done before it is refilled
  }

  // finalize: denominator = Lacc + log(sum_k w_k * exp(end_k))
  num += s_end[prev];
  const float wk[T_TAGS] = {wv[0].x, wv[0].y, wv[1].x, wv[1].y,
                            wv[2].x, wv[2].y, w6};
  float acc = wk[0] * __expf(s_end[0]);
#pragma unroll
  for (int k = 1; k < T_TAGS; ++k)
    acc = __builtin_fmaf(wk[k], __expf(s_end[k]), acc);
  const float den = Lacc + __logf(acc);
  float v = num - den;

  // deterministic block reduction
#pragma unroll
  for (int o = 16; o > 0; o >>= 1) v += __shfl_xor(v, o, 32);
  const int lane = tid & 31, wid = tid >> 5;
  if (lane == 0) s_red[wid] = v;
  __syncthreads();
  if (tid == 0) {
    float sum = 0.f;
#pragma unroll
    for (int i = 0; i < BLOCK / 32; ++i) sum += s_red[i];
    partial[blockIdx.x] = sum;
  }
}

__global__ void crf_reduce_kernel(const float* __restrict__ partial,
                                  float* __restrict__ out, int n) {
  float v = (threadIdx.x < n) ? partial[threadIdx.x] : 0.f;
#pragma unroll
  for (int o = 16; o > 0; o >>= 1) v += __shfl_xor(v, o, 32);
  if (threadIdx.x == 0) out[0] = v;
}

extern "C" void kernel_launch(void* const* d_in, const int* in_sizes, int n_in,
                              void* d_out, int out_size, void* d_ws, size_t ws_size,
                              hipStream_t stream) {
  (void)in_sizes; (void)n_in; (void)out_size; (void)ws_size;
  const float* emis   = (const float*)d_in[0];
  const float* startT = (const float*)d_in[1];
  const float* endT   = (const float*)d_in[2];
  const float* trans  = (const float*)d_in[3];
  const int*   tags   = (const int*)d_in[4];
  const unsigned char* mask = (const unsigned char*)d_in[5];  // bool mask, 1B

  float* partial = (float*)d_ws;                 // 32 block partials
  const int nblk = B_SZ / BLOCK;                 // 32 blocks x 256 threads

  crf_fwd_kernel<<<nblk, BLOCK, 0, stream>>>(emis, startT, endT, trans, tags,
                                             mask, partial);
  crf_reduce_kernel<<<1, 32, 0, stream>>>(partial, (float*)d_out, nblk);
}